// SD2M_26886495273234
// MI455X (gfx1250) — compile-verified
//
#include <hip/hip_runtime.h>
#include <math.h>

#define N 512
#define BATCH 8
#define CH 32
#define NIMG (BATCH * CH)
#define NSEG 257   // rings 0..256

typedef float v2f __attribute__((ext_vector_type(2)));
typedef float v4f __attribute__((ext_vector_type(4)));
typedef float v8f __attribute__((ext_vector_type(8)));

// ---- CDNA5 WMMA: D(16x16,f32) = A(16x4,f32) * B(4x16,f32) + C ----
__device__ __forceinline__ v8f wmma4(v2f a, v2f b, v8f c) {
    return __builtin_amdgcn_wmma_f32_16x16x4_f32(
        /*neg_a=*/false, a, /*neg_b=*/false, b,
        /*c_mod=*/(short)0, c, /*reuse_a=*/false, /*reuse_b=*/false);
}

// Shifted-coordinate ring component along one axis.
__device__ __forceinline__ int ring1(int u) {
    int i = (u + N / 2) & (N - 1);
    return max(N / 2 - i, i - N / 2 + 1);
}
__device__ __forceinline__ int ring_of(int u, int v) {
    return max(ring1(u), ring1(v));
}

// DFT matrices. W[n][k] = exp(-2*pi*i*n*k/N) (symmetric in n,k).
// wr = cos, wiF = -sin (forward), wiI = +sin (inverse/conjugate).
__global__ void gen_dft_kernel(float* __restrict__ wr,
                               float* __restrict__ wiF,
                               float* __restrict__ wiI) {
    int idx = blockIdx.x * blockDim.x + threadIdx.x;   // exactly N*N threads
    int n = idx >> 9, k = idx & (N - 1);
    int nk = (n * k) & (N - 1);                        // mod N keeps precision
    float t = (-6.283185307179586f / (float)N) * (float)nk;
    float s, c;
    __sincosf(t, &s, &c);
    wr[idx] = c;
    wiF[idx] = s;     // -sin(2*pi*nk/N)
    wiI[idx] = -s;    // +sin(2*pi*nk/N)
}

__global__ void zero_ring_kernel(float* __restrict__ ring) {
    int i = blockIdx.x * blockDim.x + threadIdx.x;
    if (i < BATCH * NSEG) ring[i] = 0.f;
}

// One complex matmul pass: Y = transpose(X * W), per image.
// Template specialization keeps the K-loop straight-line (no runtime flags).
//  HAS_IMAG : input has an imaginary component
//  DO_MASK  : zero input elements with ring(u,v) <= cutoff[b] (fused)
//  WRITE_ABS: final pass, store |y|*scale into Yabs instead of (Yr,Yi)
template <bool HAS_IMAG, bool DO_MASK, bool WRITE_ABS>
__global__ void fft_pass_kernel(const float* __restrict__ Xr,
                                const float* __restrict__ Xi,
                                const float* __restrict__ Wr,
                                const float* __restrict__ Wi,
                                float scale,
                                float* __restrict__ Yr,
                                float* __restrict__ Yi,
                                float* __restrict__ Yabs,
                                const int* __restrict__ cut)
{
    const int lane = threadIdx.x & 31;
    const int wave = threadIdx.x >> 5;
    const int tn   = blockIdx.x * 4 + wave;   // 16-wide output column tile
    const int tm   = blockIdx.y;              // 16-tall output row tile
    const int img  = blockIdx.z;
    const size_t ibase = (size_t)img * N * N;

    // A fragment (16x4 f32): lane holds M = lane&15, K = kOff, kOff+1
    const int aM   = lane & 15;
    const int kOff = (lane >> 4) * 2;
    // B fragment (4x16 f32): lane holds N = lane&15, K = kOff, kOff+1
    const int bN   = lane & 15;

    const int rowA = tm * 16 + aM;   // u coordinate of this A row
    const int colB = tn * 16 + bN;

    int cutoff = 0, ringRow = 0;
    if (DO_MASK) {
        cutoff  = cut[img >> 5];     // img = b*CH + c
        ringRow = ring1(rowA);       // row component, constant per thread
    }

    const float* xrp = Xr + ibase + (size_t)rowA * N;
    const float* xip = HAS_IMAG ? (Xi + ibase + (size_t)rowA * N) : nullptr;
    // Symmetry W[k][colB] == W[colB][k]: read B fragments contiguously.
    const float* wrp = Wr + (size_t)colB * N;
    const float* wip = Wi + (size_t)colB * N;

    v8f accRR = {}; v8f accII = {}; v8f accRI = {}; v8f accIR = {};

#pragma unroll 4
    for (int k0 = 0; k0 < N; k0 += 4) {
        const int ka = k0 + kOff;
        v2f ar = *(const v2f*)(xrp + ka);
        v2f ai;
        if (HAS_IMAG) ai = *(const v2f*)(xip + ka);
        if (DO_MASK) {
            float m0 = (max(ringRow, ring1(ka))     > cutoff) ? 1.f : 0.f;
            float m1 = (max(ringRow, ring1(ka + 1)) > cutoff) ? 1.f : 0.f;
            ar.x *= m0; ar.y *= m1;
            if (HAS_IMAG) { ai.x *= m0; ai.y *= m1; }
        }
        const v2f br = *(const v2f*)(wrp + ka);
        const v2f bi = *(const v2f*)(wip + ka);

        accRR = wmma4(ar, br, accRR);
        accRI = wmma4(ar, bi, accRI);
        if (HAS_IMAG) {
            accII = wmma4(ai, bi, accII);
            accIR = wmma4(ai, br, accIR);
        }
    }

    // C/D layout: VGPR v holds M = v + 8*(lane>=16), N = lane&15.
    // Per lane the 8 results are CONTIGUOUS in the transposed output row:
    // o(v) = ibase + (tn*16 + nLocal)*N + tm*16 + mBase + v  -> two b128 stores.
    const int nLocal = lane & 15;
    const int mBase  = (lane >> 4) * 8;
    const size_t obase = ibase + (size_t)(tn * 16 + nLocal) * N + (tm * 16 + mBase);

    float yr[8], yi[8];
#pragma unroll
    for (int v = 0; v < 8; v++) {
        if (HAS_IMAG) {
            yr[v] = (accRR[v] - accII[v]) * scale;
            yi[v] = (accRI[v] + accIR[v]) * scale;
        } else {
            yr[v] = accRR[v] * scale;
            yi[v] = accRI[v] * scale;
        }
    }

    if (WRITE_ABS) {
        float ya[8];
#pragma unroll
        for (int v = 0; v < 8; v++) ya[v] = sqrtf(yr[v] * yr[v] + yi[v] * yi[v]);
        *(v4f*)(Yabs + obase)     = *(v4f*)&ya[0];
        *(v4f*)(Yabs + obase + 4) = *(v4f*)&ya[4];
    } else {
        *(v4f*)(Yr + obase)     = *(v4f*)&yr[0];
        *(v4f*)(Yr + obase + 4) = *(v4f*)&yr[4];
        *(v4f*)(Yi + obase)     = *(v4f*)&yi[0];
        *(v4f*)(Yi + obase + 4) = *(v4f*)&yi[4];
    }
}

// Ring-energy histogram of channel 0 per batch. One block per (row u, batch b),
// LDS-staged atomics, then flush to global bins.
__global__ void ring_energy_kernel(const float* __restrict__ Fr,
                                   const float* __restrict__ Fi,
                                   float* __restrict__ ring)
{
    __shared__ float h[NSEG];
    const int u = blockIdx.x;
    const int b = blockIdx.y;
    for (int s = threadIdx.x; s < NSEG; s += blockDim.x) h[s] = 0.f;
    __syncthreads();
    const size_t base = (size_t)(b * CH) * N * N + (size_t)u * N;  // channel 0
    const int ru = ring1(u);
    for (int v = threadIdx.x; v < N; v += blockDim.x) {
        float fr = Fr[base + v], fi = Fi[base + v];
        atomicAdd(&h[max(ru, ring1(v))], fr * fr + fi * fi);
    }
    __syncthreads();
    for (int s = threadIdx.x; s < NSEG; s += blockDim.x)
        if (h[s] != 0.f) atomicAdd(&ring[b * NSEG + s], h[s]);
}

// Per-batch cumulative sum + 50% energy cutoff. cand over rings 1..255.
__global__ void cutoff_kernel(const float* __restrict__ ring,
                              int* __restrict__ cut)
{
    int b = threadIdx.x;
    if (b >= BATCH) return;
    float total = 0.f;
    for (int s = 0; s < NSEG; s++) total += ring[b * NSEG + s];
    const float target = total * 0.5f;
    float cum = ring[b * NSEG + 0];
    int first = 0;
    for (int s = 1; s < N / 2; s++) {      // s = 1 .. 255
        cum += ring[b * NSEG + s];
        if (first == 0 && cum >= target) first = s;
    }
    cut[b] = (first != 0) ? first : 5;
}

extern "C" void kernel_launch(void* const* d_in, const int* in_sizes, int n_in,
                              void* d_out, int out_size, void* d_ws, size_t ws_size,
                              hipStream_t stream) {
    const float* x = (const float*)d_in[0];
    float* out = (float*)d_out;
    float* ws = (float*)d_ws;

    const size_t imgsz = (size_t)N * N;        // 262,144 floats (1 MB)
    const size_t bufsz = (size_t)NIMG * imgsz; // 67,108,864 floats (256 MB)

    float* wr   = ws;
    float* wiF  = wr + imgsz;
    float* wiI  = wiF + imgsz;
    float* Ar   = wiI + imgsz;
    float* Ai   = Ar + bufsz;
    float* Br   = Ai + bufsz;
    float* Bi   = Br + bufsz;
    float* ring = Bi + bufsz;
    int*   cut  = (int*)(ring + BATCH * NSEG);

    // Build DFT matrices (L2-resident: 3 MB).
    gen_dft_kernel<<<dim3((N * N) / 256), dim3(256), 0, stream>>>(wr, wiF, wiI);
    zero_ring_kernel<<<dim3((BATCH * NSEG + 255) / 256), dim3(256), 0, stream>>>(ring);

    const dim3 pgrid(N / 16 / 4, N / 16, NIMG);  // (8, 32, 256), 4 waves/block
    const dim3 pblock(128);
    const float inv = 1.0f / (float)N;

    // Forward FFT: two transposing row passes. Pass 1 input is purely real.
    fft_pass_kernel<false, false, false><<<pgrid, pblock, 0, stream>>>(
        x, nullptr, wr, wiF, 1.f, Ar, Ai, nullptr, nullptr);
    fft_pass_kernel<true, false, false><<<pgrid, pblock, 0, stream>>>(
        Ar, Ai, wr, wiF, 1.f, Br, Bi, nullptr, nullptr);

    // Ring energy of channel 0, per batch; then cutoff.
    ring_energy_kernel<<<dim3(N, BATCH), dim3(256), 0, stream>>>(Br, Bi, ring);
    cutoff_kernel<<<dim3(1), dim3(32), 0, stream>>>(ring, cut);

    // Inverse FFT: mask fused into pass 1 loads; pass 2 emits |y| directly.
    fft_pass_kernel<true, true, false><<<pgrid, pblock, 0, stream>>>(
        Br, Bi, wr, wiI, inv, Ar, Ai, nullptr, cut);
    fft_pass_kernel<true, false, true><<<pgrid, pblock, 0, stream>>>(
        Ar, Ai, wr, wiI, inv, nullptr, nullptr, out, nullptr);
}